// PowerSpectrum_38104949850574
// MI455X (gfx1250) — compile-verified
//
#include <hip/hip_runtime.h>
#include <math.h>

typedef __attribute__((ext_vector_type(2))) float v2f;
typedef __attribute__((ext_vector_type(4))) float v4f;
typedef __attribute__((ext_vector_type(8))) float v8f;

#define LSH        7
#define NN         32           // N = NSP*NMAX
#define MCOL       49           // LSH*LSH
#define ROW_F      1568         // NN*MCOL floats per sample
#define OUT_F      3808         // 544 padded pairs * 7
#define OUT_VALID  3696         // 528 pairs * 7  (== 924 float4s exactly)
#define WPB        2            // waves (=samples) per block
#define SAMP_F     (ROW_F + OUT_F)  // 5376 floats of LDS per sample

__device__ __forceinline__ v8f wmma_f32_16x16x4(v2f a, v2f b, v8f c) {
#if defined(__gfx1250__)
    return __builtin_amdgcn_wmma_f32_16x16x4_f32(
        /*neg_a=*/false, a, /*neg_b=*/false, b,
        /*c_mod=*/(short)0, c, /*reuse_a=*/false, /*reuse_b=*/false);
#else
    (void)a; (void)b;
    return c;
#endif
}

__global__ __launch_bounds__(WPB * 32)
void ps_wmma_kernel(const float* __restrict__ se, float* __restrict__ out, int J) {
    __shared__ __align__(16) float lds[WPB * SAMP_F];

    const int lane   = threadIdx.x & 31;
    const int w      = threadIdx.x >> 5;
    const int sample = blockIdx.x * WPB + w;
    if (sample >= J) return;                // wave-uniform -> EXEC stays all-1s

    float* sev = &lds[w * SAMP_F];          // 32x49 fp32 tile (stride 49: gcd(49,64)=1, bank-clean)
    float* stg = sev + ROW_F;               // 3808-float output staging

    // ---- streamed, vectorized load into LDS (same-wave ordering, no barrier)
    {
        const float* src = se + (size_t)sample * ROW_F;
        const v4f*   s4  = (const v4f*)src;
        v4f*         d4  = (v4f*)sev;
        #pragma unroll
        for (int it = 0; it < 12; ++it) {   // 12*128 = 1536 floats
            const int i4 = it * 32 + lane;
            d4[i4] = __builtin_nontemporal_load(s4 + i4);
        }
        sev[1536 + lane] = __builtin_nontemporal_load(src + 1536 + lane); // tail 32
    }

    const int   half = lane >> 4;           // 0: lanes 0-15, 1: lanes 16-31
    const int   r    = lane & 15;
    const int   row0 = r * MCOL;            // fragment rows 0..15
    const int   row1 = (16 + r) * MCOL;     // fragment rows 16..31
    const float SQ2  = 1.41421356237309515f;

    // ---- precompute l-invariant staging addresses & diagonal factors (once)
    int   s00[8], s01[8], s11[8];
    float fac[8];
    #pragma unroll
    for (int vg = 0; vg < 8; ++vg) {
        const int  p    = vg + 8 * half;    // C/D row within tile (0..15)
        const int  q    = r;                // C/D col within tile (0..15)
        const bool cond = (p <= q);
        fac[vg] = (p == q) ? 1.0f : SQ2;
        const int t01 = p * NN - (p * (p - 1)) / 2 + (16 + q - p);  // pair (p, 16+q): always valid
        const int t00 = t01 - 16;                                    // pair (p, q)
        const int p2  = p + 16;
        const int t11 = p2 * NN - (p2 * (p2 - 1)) / 2 + (q - p);    // pair (p+16, q+16)
        s01[vg] = t01 * LSH;
        // invalid triangle entries redirected into pad region (overwritten with 0 at write-out)
        s00[vg] = cond ? t00 * LSH : (OUT_VALID + vg);
        s11[vg] = cond ? t11 * LSH : (OUT_VALID + 32 + vg);
    }

    float sumsq = 0.0f;

    #pragma unroll
    for (int l = 0; l < LSH; ++l) {
        const int kb = l * l;
        const int ke = kb + 2 * l + 1;

        v8f a00 = {0,0,0,0,0,0,0,0};
        v8f a01 = {0,0,0,0,0,0,0,0};
        v8f a11 = {0,0,0,0,0,0,0,0};

        #pragma unroll
        for (int k0 = kb; k0 < ke; k0 += 4) {
            const int kk = k0 + 2 * half;   // V0 holds K{0,2}, V1 holds K{1,3}
            // unconditional loads, value-masked (folds to constants for interior chunks)
            const float x0 = sev[row0 + kk], x1 = sev[row0 + kk + 1];
            const float y0 = sev[row1 + kk], y1 = sev[row1 + kk + 1];
            v2f f0, f1;
            f0.x = (kk     < ke) ? x0 : 0.0f;
            f0.y = (kk + 1 < ke) ? x1 : 0.0f;
            f1.x = (kk     < ke) ? y0 : 0.0f;
            f1.y = (kk + 1 < ke) ? y1 : 0.0f;
            // 3 unique Gram tiles from 2 shared fragments
            a00 = wmma_f32_16x16x4(f0, f0, a00);
            a01 = wmma_f32_16x16x4(f0, f1, a01);
            a11 = wmma_f32_16x16x4(f1, f1, a11);
        }

        const float sc  = 1.0f / sqrtf((float)(2 * l + 1));
        const float sc2 = sc * SQ2;

        // norm via full-matrix symmetry: diag once, off-diag twice == fac^2 weighting
        float tss = 0.0f;
        #pragma unroll
        for (int vg = 0; vg < 8; ++vg) {
            tss += a00[vg] * a00[vg] + a11[vg] * a11[vg] + 2.0f * a01[vg] * a01[vg];
            stg[s01[vg] + l] = a01[vg] * sc2;
            stg[s00[vg] + l] = a00[vg] * (sc * fac[vg]);
            stg[s11[vg] + l] = a11[vg] * (sc * fac[vg]);
        }
        sumsq += tss * (sc * sc);
    }

    // ---- wave32 reduction, then normalized streamed write-out
    #pragma unroll
    for (int off = 16; off > 0; off >>= 1)
        sumsq += __shfl_xor(sumsq, off, 32);
    const float scale = 1.0f / fmaxf(sqrtf(sumsq), 1e-12f);

    v4f*       dst4 = (v4f*)(out + (size_t)sample * OUT_F);
    const v4f* stg4 = (const v4f*)stg;
    for (int i = lane; i < OUT_F / 4; i += 32) {      // 952 vec4 per sample
        v4f v = {0.0f, 0.0f, 0.0f, 0.0f};
        if (i < OUT_VALID / 4) {                      // 924 valid vec4, rest forced 0
            v = stg4[i];
            v *= scale;
        }
        __builtin_nontemporal_store(v, dst4 + i);
    }
}

extern "C" void kernel_launch(void* const* d_in, const int* in_sizes, int n_in,
                              void* d_out, int out_size, void* d_ws, size_t ws_size,
                              hipStream_t stream) {
    (void)n_in; (void)out_size; (void)d_ws; (void)ws_size;
    const float* se  = (const float*)d_in[0];
    float*       out = (float*)d_out;
    const int J      = in_sizes[0] / ROW_F;           // 20000
    const int blocks = (J + WPB - 1) / WPB;
    ps_wmma_kernel<<<blocks, WPB * 32, 0, stream>>>(se, out, J);
}